// TransR_85727547228324
// MI455X (gfx1250) — compile-verified
//
#include <hip/hip_runtime.h>

// CDNA5 (gfx1250) wave32 WMMA types
typedef __attribute__((ext_vector_type(2))) float v2f;
typedef __attribute__((ext_vector_type(8))) float v8f;

#define D 128
#define KSTEPS (D / 4)          // 32 WMMA k-steps of K=4
#define WAVES_PER_BLOCK 4

// One wave per sample b.  D = A(16x4 fp32) x B(4x16 fp32) + C, V_WMMA_F32_16X16X4_F32.
// A rows 0..2 = h_e / pos_e / neg_e (rows 3..15 zero), held entirely in VGPRs
// (v2f aR[32] = 64 VGPRs), loaded once from LDS and reused across all 8 column tiles.
//   A layout: lane L<16 -> (M=L, K=k0+0 / k0+1); lanes 16..31 -> K=k0+2 / k0+3.
//   B layout: lane L<16 -> (K=k0+0/1, N=n0+L);   lanes 16..31 -> K=k0+2/3.
//   C layout: VGPR i, lanes 0..15 => row M=i  (rows 0/1/2 = trans_h/pos/neg).
__global__ __launch_bounds__(128) void TransR_wmma_kernel(
    const int* __restrict__ h,
    const int* __restrict__ r,
    const int* __restrict__ pos_t,
    const int* __restrict__ neg_t,
    const float* __restrict__ entity_embed,
    const float* __restrict__ relation_embed,
    const float* __restrict__ trans_M,
    float* __restrict__ out,
    int B)
{
    __shared__ float vecs[WAVES_PER_BLOCK][3][D];   // h_e / pos_e / neg_e per wave

    const int lane = threadIdx.x & 31;
    const int wave = threadIdx.x >> 5;
    const int b    = blockIdx.x * WAVES_PER_BLOCK + wave;
    const bool active = (b < B);

    int hv = 0, rv = 0, pv = 0, nv = 0;
    if (active) {
        hv = h[b];
        rv = r[b];
        pv = pos_t[b];
        nv = neg_t[b];
    }

    // Stage the 3 gathered entity vectors into this wave's LDS region (float4 wide:
    // rows are 512B-aligned, one b128 per lane per vector).
    if (active) {
        const float4* __restrict__ ev = (const float4*)entity_embed;
        float4* __restrict__ dst = (float4*)&vecs[wave][0][0];
        dst[0 * 32 + lane] = ev[hv * 32 + lane];
        dst[1 * 32 + lane] = ev[pv * 32 + lane];
        dst[2 * 32 + lane] = ev[nv * 32 + lane];
    }
    __syncthreads();

    if (active) {
        const float* __restrict__ W = trans_M + rv * (D * D);
        float* __restrict__ out_h = out;                // trans_h
        float* __restrict__ out_r = out + 1 * B * D;    // r_e
        float* __restrict__ out_p = out + 2 * B * D;    // trans_pos
        float* __restrict__ out_n = out + 3 * B * D;    // trans_neg

        // r_e gather-copy, b128 wide (4 floats per lane covers D=128)
        {
            const float4* __restrict__ rvrow = (const float4*)(relation_embed + rv * D);
            float4* __restrict__ rdst = (float4*)(out_r + b * D);
            rdst[lane] = rvrow[lane];
        }

        const int row   = lane & 15;            // M (A) / N offset (B)
        const int khalf = (lane >> 4) << 1;     // 0 (lanes 0-15) or 2 (lanes 16-31)
        const int srow  = (row < 3) ? row : 0;  // clamp for dead A rows
        const bool rowValid = (row < 3);

        // Load the whole A operand into registers ONCE (64 VGPRs), zero dead rows once.
        v2f aR[KSTEPS];
        {
            const float* __restrict__ avec = &vecs[wave][srow][0];
            #pragma unroll
            for (int kk = 0; kk < KSTEPS; ++kk) {
                const float ax = avec[kk * 4 + khalf];        // ds_load_b64 pair
                const float ay = avec[kk * 4 + khalf + 1];
                aR[kk].x = rowValid ? ax : 0.0f;
                aR[kk].y = rowValid ? ay : 0.0f;
            }
        }

        for (int n0 = 0; n0 < D; n0 += 16) {    // 8 output column tiles
            v8f c = {};
            const float* __restrict__ Wc = W + n0 + row;   // one base, imm offsets below
            #pragma unroll
            for (int kk = 0; kk < KSTEPS; ++kk) {          // 32 fp32 WMMAs
                v2f bm;
                bm.x = Wc[(kk * 4 + khalf) * D];
                bm.y = Wc[(kk * 4 + khalf + 1) * D];
                c = __builtin_amdgcn_wmma_f32_16x16x4_f32(
                        /*neg_a=*/false, aR[kk], /*neg_b=*/false, bm,
                        /*c_mod=*/(short)0, c, /*reuse_a=*/false, /*reuse_b=*/false);
            }
            if (lane < 16) {                    // rows M=0,1,2 of C
                const int o = b * D + n0 + lane;
                out_h[o] = c[0];
                out_p[o] = c[1];
                out_n[o] = c[2];
            }
        }
    }
}

extern "C" void kernel_launch(void* const* d_in, const int* in_sizes, int n_in,
                              void* d_out, int out_size, void* d_ws, size_t ws_size,
                              hipStream_t stream) {
    (void)n_in; (void)out_size; (void)d_ws; (void)ws_size;
    const int*   h_idx   = (const int*)d_in[0];
    const int*   r_idx   = (const int*)d_in[1];
    const int*   pos_idx = (const int*)d_in[2];
    const int*   neg_idx = (const int*)d_in[3];
    const float* ent     = (const float*)d_in[4];
    const float* rel     = (const float*)d_in[5];
    const float* transM  = (const float*)d_in[6];
    float*       out     = (float*)d_out;

    const int B = in_sizes[0];
    const int grid = (B + WAVES_PER_BLOCK - 1) / WAVES_PER_BLOCK;
    TransR_wmma_kernel<<<grid, 32 * WAVES_PER_BLOCK, 0, stream>>>(
        h_idx, r_idx, pos_idx, neg_idx, ent, rel, transM, out, B);
}